// GNNLayer_7516192768729
// MI455X (gfx1250) — compile-verified
//
#include <hip/hip_runtime.h>
#include <hip/hip_bf16.h>

typedef __attribute__((ext_vector_type(2))) float v2f;
typedef __attribute__((ext_vector_type(8))) float v8f;

#define D 128
#define PAD_WN 132   // 128 + 4 : bank-conflict-free stride (132 % 64 == 4)
#define PAD_WE 260   // 256 + 4
#define PAD_WU 388   // 384 + 4

__device__ __forceinline__ float gelu_exact(float v) {
    // exact (erf) GELU, matches jax.nn.gelu(approximate=False)
    return 0.5f * v * (1.0f + erff(v * 0.70710678118654752f));
}

// Cooperatively stage a row-major weight matrix [rows x cols] into padded LDS.
__device__ __forceinline__ void load_w_lds(float* lds, const float* __restrict__ W,
                                           int rows, int cols, int pad) {
    const int nv4 = cols >> 2;
    const int total = rows * nv4;
    for (int idx = threadIdx.x; idx < total; idx += blockDim.x) {
        const int r  = idx / nv4;
        const int c4 = (idx - r * nv4) << 2;
        const float4 v = *(const float4*)(W + r * cols + c4);
        *(float4*)(lds + r * pad + c4) = v;
    }
}

// ---------------------------------------------------------------------------
// Kernel 1: node_emb = x @ Wn^T + bn          [N,128] x [128,128]
// One wave computes a 16x128 output tile with V_WMMA_F32_16X16X4_F32.
// ---------------------------------------------------------------------------
__global__ void node_linear_kernel(const float* __restrict__ x,
                                   const float* __restrict__ Wn,
                                   const float* __restrict__ bn,
                                   float* __restrict__ node_emb,
                                   int nTiles) {
    extern __shared__ float lds[];                 // [128][PAD_WN]
    load_w_lds(lds, Wn, 128, 128, PAD_WN);
    __syncthreads();

    const int lane = threadIdx.x & 31;
    const int tile = blockIdx.x * (blockDim.x >> 5) + (threadIdx.x >> 5);
    if (tile >= nTiles) return;                    // uniform per wave -> EXEC stays all-ones

    const int m0    = tile << 4;
    const int mlow  = lane & 15;
    const int hi    = lane >> 4;
    const int khalf = hi << 1;                     // lanes 16..31 hold K+2,K+3 (ISA A layout)

    const float* ap    = x + (m0 + mlow) * D + khalf;
    const float* bbase = lds + mlow * PAD_WN + khalf;   // B[k][n] = Wn[n][k]

    v8f acc[8] = {};
    for (int k = 0; k < 128; k += 4) {
        const v2f a = *(const v2f*)(ap + k);
        #pragma unroll
        for (int t = 0; t < 8; ++t) {
            const v2f b = *(const v2f*)(bbase + t * (16 * PAD_WN) + k);
            acc[t] = __builtin_amdgcn_wmma_f32_16x16x4_f32(
                false, a, false, b, (short)0, acc[t], false, false);
        }
    }

    #pragma unroll
    for (int t = 0; t < 8; ++t) {
        const int n = (t << 4) + mlow;
        const float bv = bn[n];
        #pragma unroll
        for (int r = 0; r < 8; ++r)                // C layout: VGPR r -> row r (+8 for hi lanes)
            node_emb[(m0 + r + (hi << 3)) * D + n] = acc[t][r] + bv;
    }
}

// ---------------------------------------------------------------------------
// Kernel 2: per 16-edge tile:
//   edge_emb = x[src] @ WeL^T + x[dst] @ WeR^T + be   (gathered-A WMMA GEMM)
//   agg_edge[src] += edge_emb          (f32 atomics)
//   agg_node[src] += node_emb[dst]     (f32 atomics)
// ---------------------------------------------------------------------------
__global__ void edge_kernel(const float* __restrict__ x,
                            const int* __restrict__ srcI,
                            const int* __restrict__ dstI,
                            const float* __restrict__ We,
                            const float* __restrict__ be,
                            const float* __restrict__ node_emb,
                            float* __restrict__ agg_node,
                            float* __restrict__ agg_edge,
                            int eTiles) {
    extern __shared__ float lds[];                 // [128][PAD_WE], We row-major (n,k)
    load_w_lds(lds, We, 128, 256, PAD_WE);
    __syncthreads();

    const int lane = threadIdx.x & 31;
    const int tile = blockIdx.x * (blockDim.x >> 5) + (threadIdx.x >> 5);
    if (tile >= eTiles) return;

    const int e0    = tile << 4;
    const int mlow  = lane & 15;
    const int hi    = lane >> 4;
    const int khalf = hi << 1;

    const int sIdx = srcI[e0 + mlow];              // gather row ids (x is L2-resident)
    const int dIdx = dstI[e0 + mlow];
    const float* aps   = x + sIdx * D + khalf;
    const float* apd   = x + dIdx * D + khalf;
    const float* bbase = lds + mlow * PAD_WE + khalf;

    v8f acc[8] = {};
    for (int k = 0; k < 128; k += 4) {             // left half: x[src] @ We[:, :128]^T
        const v2f a = *(const v2f*)(aps + k);
        #pragma unroll
        for (int t = 0; t < 8; ++t) {
            const v2f b = *(const v2f*)(bbase + t * (16 * PAD_WE) + k);
            acc[t] = __builtin_amdgcn_wmma_f32_16x16x4_f32(
                false, a, false, b, (short)0, acc[t], false, false);
        }
    }
    for (int k = 0; k < 128; k += 4) {             // right half: x[dst] @ We[:, 128:]^T
        const v2f a = *(const v2f*)(apd + k);
        #pragma unroll
        for (int t = 0; t < 8; ++t) {
            const v2f b = *(const v2f*)(bbase + t * (16 * PAD_WE) + 128 + k);
            acc[t] = __builtin_amdgcn_wmma_f32_16x16x4_f32(
                false, a, false, b, (short)0, acc[t], false, false);
        }
    }

    // scatter edge_emb into agg_edge[src]
    int srows[8];
    #pragma unroll
    for (int r = 0; r < 8; ++r) srows[r] = srcI[e0 + (hi << 3) + r];
    #pragma unroll
    for (int t = 0; t < 8; ++t) {
        const int n = (t << 4) + mlow;
        const float bv = be[n];
        #pragma unroll
        for (int r = 0; r < 8; ++r)
            atomicAdd(agg_edge + srows[r] * D + n, acc[t][r] + bv);
    }

    // agg_node[src] += node_emb[dst] for the 16 edges of this tile
    #pragma unroll 1
    for (int e = 0; e < 16; ++e) {
        const int s = __shfl(sIdx, e, 32);
        const int d = __shfl(dIdx, e, 32);
        const float* nb = node_emb + d * D;
        float*       ab = agg_node + s * D;
        #pragma unroll
        for (int c = 0; c < 4; ++c)
            atomicAdd(ab + lane + c * 32, nb[lane + c * 32]);
    }
}

// ---------------------------------------------------------------------------
// Kernel 3: out = gelu([node_emb | agg_node | agg_edge]) @ Wu^T + bu
//   [N,384] x [384,128]; GELU applied on the fly to A fragments.
// ---------------------------------------------------------------------------
__global__ void update_kernel(const float* __restrict__ node_emb,
                              const float* __restrict__ agg_node,
                              const float* __restrict__ agg_edge,
                              const float* __restrict__ Wu,
                              const float* __restrict__ bu,
                              float* __restrict__ out,
                              int nTiles) {
    extern __shared__ float lds[];                 // [128][PAD_WU]
    load_w_lds(lds, Wu, 128, 384, PAD_WU);
    __syncthreads();

    const int lane = threadIdx.x & 31;
    const int tile = blockIdx.x * (blockDim.x >> 5) + (threadIdx.x >> 5);
    if (tile >= nTiles) return;

    const int m0    = tile << 4;
    const int mlow  = lane & 15;
    const int hi    = lane >> 4;
    const int khalf = hi << 1;

    const float* bbase = lds + mlow * PAD_WU + khalf;
    const float* parts[3] = { node_emb, agg_node, agg_edge };

    v8f acc[8] = {};
    #pragma unroll 1
    for (int part = 0; part < 3; ++part) {
        const float* ap  = parts[part] + (m0 + mlow) * D + khalf;
        const int    kg0 = part * 128;             // column offset within Wu's K=384
        for (int k = 0; k < 128; k += 4) {
            const v2f raw = *(const v2f*)(ap + k);
            v2f a;
            a[0] = gelu_exact(raw[0]);
            a[1] = gelu_exact(raw[1]);
            #pragma unroll
            for (int t = 0; t < 8; ++t) {
                const v2f b = *(const v2f*)(bbase + t * (16 * PAD_WU) + kg0 + k);
                acc[t] = __builtin_amdgcn_wmma_f32_16x16x4_f32(
                    false, a, false, b, (short)0, acc[t], false, false);
            }
        }
    }

    #pragma unroll
    for (int t = 0; t < 8; ++t) {
        const int n = (t << 4) + mlow;
        const float bv = bu[n];
        #pragma unroll
        for (int r = 0; r < 8; ++r)
            out[(m0 + r + (hi << 3)) * D + n] = acc[t][r] + bv;
    }
}

// ---------------------------------------------------------------------------
extern "C" void kernel_launch(void* const* d_in, const int* in_sizes, int n_in,
                              void* d_out, int out_size, void* d_ws, size_t ws_size,
                              hipStream_t stream) {
    const float* x  = (const float*)d_in[0];
    const int*   ei = (const int*)d_in[1];
    const float* Wn = (const float*)d_in[2];
    const float* bn = (const float*)d_in[3];
    const float* We = (const float*)d_in[4];
    const float* be = (const float*)d_in[5];
    const float* Wu = (const float*)d_in[6];
    const float* bu = (const float*)d_in[7];

    const int N = in_sizes[0] / D;      // 100000 (multiple of 16)
    const int E = in_sizes[1] / 2;      // 600000 (multiple of 16)
    const int* srcI = ei;
    const int* dstI = ei + E;

    float* node_emb = (float*)d_ws;
    float* agg_node = node_emb + (size_t)N * D;
    float* agg_edge = agg_node + (size_t)N * D;

    // zero the two accumulation buffers (contiguous) — capture-safe async memset
    hipMemsetAsync(agg_node, 0, 2 * (size_t)N * D * sizeof(float), stream);

    const int wpb = 8;                              // 8 waves = 256 threads / block
    const int nTiles = (N + 15) >> 4;
    const int eTiles = (E + 15) >> 4;
    const dim3 blk(32 * wpb);

    node_linear_kernel<<<(nTiles + wpb - 1) / wpb, blk,
                         128 * PAD_WN * sizeof(float), stream>>>(
        x, Wn, bn, node_emb, nTiles);

    edge_kernel<<<(eTiles + wpb - 1) / wpb, blk,
                  128 * PAD_WE * sizeof(float), stream>>>(
        x, srcI, dstI, We, be, node_emb, agg_node, agg_edge, eTiles);

    update_kernel<<<(nTiles + wpb - 1) / wpb, blk,
                    128 * PAD_WU * sizeof(float), stream>>>(
        node_emb, agg_node, agg_edge, Wu, bu, (float*)d_out, nTiles);
}